// GCN_88321707475504
// MI455X (gfx1250) — compile-verified
//
#include <hip/hip_runtime.h>

// ---------------------------------------------------------------------------
// GCN forward for MI455X (gfx1250, wave32).  Dense GEMMs use the fp32 WMMA
// path V_WMMA_F32_16X16X4_F32 (keeps fp32 precision; SpMM traffic is the
// bottleneck, not matrix FLOPs).  SpMM via gather + global f32 atomics.
// ---------------------------------------------------------------------------

typedef float v2f __attribute__((ext_vector_type(2)));
typedef float v8f __attribute__((ext_vector_type(8)));

#define N_NODES_C 100000
#define IN_DIM_C  256
#define HID_DIM_C 128
#define OUT_DIM_C 64

// ---------------------------------------------------------------- fill zero
__global__ void k_fill_zero(float* __restrict__ p, long long n) {
  long long i = (long long)blockIdx.x * blockDim.x + threadIdx.x;
  if (i < n) p[i] = 0.0f;
}

// ---------------------------------------------------------------- degrees
__global__ void k_degrees(const int* __restrict__ src, const int* __restrict__ dst,
                          float* __restrict__ deg_out, float* __restrict__ deg_in,
                          int n_edges) {
  int e = blockIdx.x * blockDim.x + threadIdx.x;
  if (e >= n_edges) return;
  atomicAdd(&deg_out[src[e]], 1.0f);
  atomicAdd(&deg_in[dst[e]], 1.0f);
}

// --------------------------------------------------- clamp(.,1) then rsqrt
__global__ void k_rsqrt_clamp(float* __restrict__ d, int n) {
  int i = blockIdx.x * blockDim.x + threadIdx.x;
  if (i >= n) return;
  float v = d[i];
  v = v < 1.0f ? 1.0f : v;
  d[i] = rsqrtf(v);
}

// ---------------------------------------------------------------- WMMA GEMM
// Y[row, col] = sum_k (X[row,k] * rowscale[row]) * W[k, col]
// One wave computes one 16x16 output tile.  blockIdx.x = M tile (rows),
// wave-in-block = N tile (cols).  K stepped by 4 via v_wmma_f32_16x16x4_f32.
//
// A-matrix 16x4 f32 layout (ISA 7.12.2): lanes 0-15 -> M=lane, VGPR0=K0,
// VGPR1=K1; lanes 16-31 -> M=lane-16, VGPR0=K2, VGPR1=K3.  So a lane's
// float2 holds K = {2*half, 2*half+1}.  B 4x16 mirrored (N = lane&15).
// D: VGPR r -> M = r + 8*half, N = lane&15.
__global__ __launch_bounds__(256)
void k_gemm_wmma(const float* __restrict__ X, const float* __restrict__ W,
                 const float* __restrict__ rowscale, float* __restrict__ Y,
                 int K, int NC) {
  const int lane = threadIdx.x & 31;
  const int wave = threadIdx.x >> 5;          // N tile within block
  const int half = lane >> 4;                 // 0 or 1
  const int lid  = lane & 15;
  const int row  = blockIdx.x * 16 + lid;     // grid sized so always valid
  const int col  = wave * 16 + lid;

  const float s = rowscale[row];
  const float* __restrict__ Arow = X + (size_t)row * K;

  v8f acc = {};
  for (int k = 0; k < K; k += 4) {
    const int ka = k + 2 * half;
    v2f a, b;
    a.x = Arow[ka]     * s;
    a.y = Arow[ka + 1] * s;
    b.x = W[(size_t)ka       * NC + col];
    b.y = W[(size_t)(ka + 1) * NC + col];
    acc = __builtin_amdgcn_wmma_f32_16x16x4_f32(
        /*neg_a=*/false, a, /*neg_b=*/false, b,
        /*c_mod=*/(short)0, acc, /*reuse_a=*/false, /*reuse_b=*/false);
  }

  const int orow = blockIdx.x * 16 + 8 * half;
  #pragma unroll
  for (int r = 0; r < 8; ++r)
    Y[(size_t)(orow + r) * NC + col] = acc[r];
}

// ------------------------------------------------------------- SpMM scatter
// agg[dst[e], :] += feat[src[e], :].  Each thread: one edge x 4 dims.
// shift = log2(dim/4), mask = dim/4 - 1.
__global__ void k_scatter_add(const float* __restrict__ feat, float* __restrict__ agg,
                              const int* __restrict__ src, const int* __restrict__ dst,
                              int n_edges, int dim, int shift, int mask) {
  long long idx = (long long)blockIdx.x * blockDim.x + threadIdx.x;
  int e = (int)(idx >> shift);
  if (e >= n_edges) return;
  int d = ((int)idx & mask) * 4;
  const float4 v = *(const float4*)(feat + (size_t)src[e] * dim + d);
  float* out = agg + (size_t)dst[e] * dim + d;
  atomicAdd(out + 0, v.x);
  atomicAdd(out + 1, v.y);
  atomicAdd(out + 2, v.z);
  atomicAdd(out + 3, v.w);
}

// ---------------------------------------------- x = relu(agg*dinv_in + b)
__global__ void k_norm_bias_relu(float* __restrict__ agg, const float* __restrict__ dinv_in,
                                 const float* __restrict__ b, int shift, int dmask,
                                 long long total) {
  long long i = (long long)blockIdx.x * blockDim.x + threadIdx.x;
  if (i >= total) return;
  int node = (int)(i >> shift);
  int d = (int)i & dmask;
  float v = agg[i] * dinv_in[node] + b[d];
  agg[i] = v > 0.0f ? v : 0.0f;
}

// ------------------------------- logits = agg*dinv_in + b; softmax (dim 64)
// One wave32 per node; lane handles dims {lane, lane+32}.
__global__ void k_final_softmax(const float* __restrict__ agg, const float* __restrict__ dinv_in,
                                const float* __restrict__ b, float* __restrict__ out_sm,
                                float* __restrict__ out_lg, int n_nodes) {
  int gtid = blockIdx.x * blockDim.x + threadIdx.x;
  int node = gtid >> 5;
  int lane = threadIdx.x & 31;
  if (node >= n_nodes) return;
  const float s = dinv_in[node];
  const float* a = agg + (size_t)node * OUT_DIM_C;
  float l0 = a[lane]      * s + b[lane];
  float l1 = a[lane + 32] * s + b[lane + 32];

  float m = fmaxf(l0, l1);
  #pragma unroll
  for (int o = 16; o > 0; o >>= 1) m = fmaxf(m, __shfl_xor(m, o, 32));
  float e0 = expf(l0 - m), e1 = expf(l1 - m);
  float sum = e0 + e1;
  #pragma unroll
  for (int o = 16; o > 0; o >>= 1) sum += __shfl_xor(sum, o, 32);
  float inv = 1.0f / sum;

  size_t base = (size_t)node * OUT_DIM_C;
  out_lg[base + lane]      = l0;
  out_lg[base + lane + 32] = l1;
  out_sm[base + lane]      = e0 * inv;
  out_sm[base + lane + 32] = e1 * inv;
}

// ---------------------------------------------------------------------------
extern "C" void kernel_launch(void* const* d_in, const int* in_sizes, int n_in,
                              void* d_out, int out_size, void* d_ws, size_t ws_size,
                              hipStream_t stream) {
  const float* h   = (const float*)d_in[0];   // [N, 256]
  const float* W1  = (const float*)d_in[1];   // [256, 128]
  const float* b1  = (const float*)d_in[2];   // [128]
  const float* W2  = (const float*)d_in[3];   // [128, 64]
  const float* b2  = (const float*)d_in[4];   // [64]
  const int*   src = (const int*)d_in[5];     // [E]
  const int*   dst = (const int*)d_in[6];     // [E]

  const int N = in_sizes[0] / IN_DIM_C;       // 100000 (divisible by 16)
  const int E = in_sizes[5];                  // 1600000

  // ---- workspace layout (floats) --------------------------------------
  float* ws       = (float*)d_ws;
  float* deg_out  = ws;                                  // N   -> dinv_out
  float* deg_in   = deg_out + N;                         // N   -> dinv_in
  float* agg1     = deg_in + N;                          // N*128 (zeroed)
  float* agg2     = agg1 + (size_t)N * HID_DIM_C;        // N*64  (zeroed)
  float* Y1       = agg2 + (size_t)N * OUT_DIM_C;        // N*128
  float* Y2       = Y1 + (size_t)N * HID_DIM_C;          // N*64

  float* out_softmax = (float*)d_out;                    // [N,64]
  float* out_logits  = out_softmax + (size_t)N * OUT_DIM_C;

  const int B = 256;

  // 1. zero deg_out, deg_in, agg1, agg2 (contiguous prefix of ws)
  {
    long long n = (long long)N * (2 + HID_DIM_C + OUT_DIM_C);
    long long g = (n + B - 1) / B;
    k_fill_zero<<<(unsigned)g, B, 0, stream>>>(ws, n);
  }
  // 2. degrees
  k_degrees<<<(E + B - 1) / B, B, 0, stream>>>(src, dst, deg_out, deg_in, E);
  // 3. dinv = rsqrt(clamp(deg,1))  (deg_out & deg_in are contiguous: 2N)
  k_rsqrt_clamp<<<(2 * N + B - 1) / B, B, 0, stream>>>(deg_out, 2 * N);

  // 4. Y1 = (h * dinv_out) @ W1   [N,256]x[256,128]; 8 waves/block (N tiles)
  k_gemm_wmma<<<N / 16, 8 * 32, 0, stream>>>(h, W1, deg_out, Y1, IN_DIM_C, HID_DIM_C);

  // 5. agg1[dst] += Y1[src]   (dim 128 -> 32 threads/edge, 4 dims each)
  {
    long long t = (long long)E * 32;
    k_scatter_add<<<(unsigned)((t + B - 1) / B), B, 0, stream>>>(
        Y1, agg1, src, dst, E, HID_DIM_C, 5, 31);
  }
  // 6. x = relu(agg1 * dinv_in + b1)  in place
  {
    long long t = (long long)N * HID_DIM_C;
    k_norm_bias_relu<<<(unsigned)((t + B - 1) / B), B, 0, stream>>>(
        agg1, deg_in, b1, 7, HID_DIM_C - 1, t);
  }

  // 7. Y2 = (x * dinv_out) @ W2   [N,128]x[128,64]; 4 waves/block
  k_gemm_wmma<<<N / 16, 4 * 32, 0, stream>>>(agg1, W2, deg_out, Y2, HID_DIM_C, OUT_DIM_C);

  // 8. agg2[dst] += Y2[src]   (dim 64 -> 16 threads/edge)
  {
    long long t = (long long)E * 16;
    k_scatter_add<<<(unsigned)((t + B - 1) / B), B, 0, stream>>>(
        Y2, agg2, src, dst, E, OUT_DIM_C, 4, 15);
  }
  // 9. logits = agg2 * dinv_in + b2; softmax -> d_out (softmax first, logits second)
  {
    long long t = (long long)N * 32;  // one wave per node
    k_final_softmax<<<(unsigned)((t + B - 1) / B), B, 0, stream>>>(
        agg2, deg_in, b2, out_softmax, out_logits, N);
  }
}